// GPRGNNConv_936302871057
// MI455X (gfx1250) — compile-verified
//
#include <hip/hip_runtime.h>
#include <stdint.h>

#define TPB  256
#define CH   96
#define NPB  32     // destination nodes per block (8 waves x 4 nodes)
#define SCAP 2048   // LDS-staged edge capacity per block (avg need ~512)

// ---------------- setup kernels ----------------

__global__ __launch_bounds__(TPB) void k_zero_i4(int4* __restrict__ p, int n4) {
  int i = blockIdx.x * TPB + threadIdx.x;
  if (i < n4) p[i] = make_int4(0, 0, 0, 0);
}

__global__ __launch_bounds__(TPB) void k_deg(const int* __restrict__ dst,
                                             int* __restrict__ degi, int nE) {
  int i = blockIdx.x * TPB + threadIdx.x;
  if (i < nE) atomicAdd(&degi[dst[i]], 1);
}

// single-block exclusive scan: rowptr[0..n] from degi[0..n-1]
__global__ __launch_bounds__(1024) void k_scan(const int* __restrict__ cnt,
                                               int* __restrict__ rowptr, int n) {
  __shared__ int s[1024];
  __shared__ int carry;
  if (threadIdx.x == 0) carry = 0;
  __syncthreads();
  for (int base = 0; base < n; base += 1024) {
    int i = base + (int)threadIdx.x;
    int v = (i < n) ? cnt[i] : 0;
    s[threadIdx.x] = v;
    __syncthreads();
    for (int off = 1; off < 1024; off <<= 1) {
      int t = (threadIdx.x >= (unsigned)off) ? s[threadIdx.x - off] : 0;
      __syncthreads();
      s[threadIdx.x] += t;
      __syncthreads();
    }
    int incl = s[threadIdx.x];
    if (i < n) rowptr[i] = carry + (incl - v);
    __syncthreads();                    // everyone has read carry
    if (threadIdx.x == 1023) carry += s[1023];
    __syncthreads();
  }
  if (threadIdx.x == 0) rowptr[n] = carry;
}

__global__ __launch_bounds__(TPB) void k_dinv(const int* __restrict__ degi,
                                              float* __restrict__ dinv, int n) {
  int i = blockIdx.x * TPB + threadIdx.x;
  if (i < n) {
    int d = degi[i];
    dinv[i] = (d > 0) ? rsqrtf((float)d) : 0.f;
  }
}

// place each edge into its dst row: col[p]=src, val[p]=dinv[src]*dinv[dst]
__global__ __launch_bounds__(TPB) void k_fill(const int* __restrict__ src,
                                              const int* __restrict__ dst,
                                              const int* __restrict__ rowptr,
                                              int* __restrict__ cursor,
                                              const float* __restrict__ dinv,
                                              int* __restrict__ col,
                                              float* __restrict__ val, int nE) {
  int i = blockIdx.x * TPB + threadIdx.x;
  if (i < nE) {
    int s = src[i], d = dst[i];
    int p = rowptr[d] + atomicAdd(&cursor[d], 1);
    col[p] = s;
    val[p] = dinv[s] * dinv[d];
  }
}

__global__ __launch_bounds__(TPB) void k_scale(const float4* __restrict__ x,
                                               float4* __restrict__ out,
                                               const float* __restrict__ gamma, int n4) {
  int i = blockIdx.x * TPB + threadIdx.x;
  float g = gamma[0];
  if (i < n4) {
    float4 v = x[i];
    out[i] = make_float4(g * v.x, g * v.y, g * v.z, g * v.w);
  }
}

// ---------------- fused propagation + gamma accumulation ----------------
// Block: 8 waves, 32 dst nodes. The block's CSR edge segment is contiguous:
// async-stage it into LDS (CDNA5 ASYNCcnt path), then each wave owns one dst
// node at a time, accumulating 96 channels in 3 VGPRs/lane: zero atomics,
// each hnext/out element written exactly once.

__global__ __launch_bounds__(TPB) void k_prop(const float* __restrict__ h,
                                              float* __restrict__ hn,
                                              float* __restrict__ out,
                                              const int* __restrict__ rowptr,
                                              const int* __restrict__ col,
                                              const float* __restrict__ val,
                                              const float* __restrict__ gamma,
                                              int gi, int nN, int writeH) {
  __shared__ int   s_col[SCAP];
  __shared__ float s_val[SCAP];
  __shared__ int   s_rp[NPB + 1];

  const int t    = threadIdx.x;
  const int base = blockIdx.x * NPB;

  if (t <= NPB) s_rp[t] = rowptr[min(base + t, nN)];
  __syncthreads();

  const int rs  = s_rp[0];
  const int len = s_rp[NPB] - rs;
  const bool staged = (len <= SCAP);

  if (staged) {
    for (int i = t; i < len; i += TPB) {
      unsigned lc  = (unsigned)(uintptr_t)&s_col[i];
      unsigned lv  = (unsigned)(uintptr_t)&s_val[i];
      unsigned off = (unsigned)(rs + i) * 4u;
      asm volatile("global_load_async_to_lds_b32 %0, %1, %2 offset:0"
                   :: "v"(lc), "v"(off), "s"((uint64_t)(uintptr_t)col) : "memory");
      asm volatile("global_load_async_to_lds_b32 %0, %1, %2 offset:0"
                   :: "v"(lv), "v"(off), "s"((uint64_t)(uintptr_t)val) : "memory");
    }
    asm volatile("s_wait_asynccnt 0x0" ::: "memory");
  }
  __syncthreads();

  const float g    = gamma[gi];
  const int   wave = t >> 5;
  const int   lane = t & 31;

  for (int w = wave; w < NPB; w += TPB / 32) {
    int n = base + w;
    if (n >= nN) break;
    float a0 = 0.f, a1 = 0.f, a2 = 0.f;

    if (staged) {
      int eb = s_rp[w] - rs, ee = s_rp[w + 1] - rs;
      for (int e = eb; e < ee; ++e) {
        if (e + 1 < ee)
          __builtin_prefetch(h + (size_t)s_col[e + 1] * CH + lane, 0, 3);
        const float* hp = h + (size_t)s_col[e] * CH;
        float nm = s_val[e];
        a0 = fmaf(hp[lane],      nm, a0);
        a1 = fmaf(hp[32 + lane], nm, a1);
        a2 = fmaf(hp[64 + lane], nm, a2);
      }
    } else {  // overflow fallback (essentially never taken: Poisson tail)
      int eb = s_rp[w], ee = s_rp[w + 1];
      for (int e = eb; e < ee; ++e) {
        const float* hp = h + (size_t)col[e] * CH;
        float nm = val[e];
        a0 = fmaf(hp[lane],      nm, a0);
        a1 = fmaf(hp[32 + lane], nm, a1);
        a2 = fmaf(hp[64 + lane], nm, a2);
      }
    }

    size_t o = (size_t)n * CH;
    if (writeH) {
      hn[o + lane]      = a0;
      hn[o + 32 + lane] = a1;
      hn[o + 64 + lane] = a2;
    }
    out[o + lane]      += g * a0;
    out[o + 32 + lane] += g * a1;
    out[o + 64 + lane] += g * a2;
  }
}

// ---------------- host driver ----------------

extern "C" void kernel_launch(void* const* d_in, const int* in_sizes, int n_in,
                              void* d_out, int out_size, void* d_ws, size_t ws_size,
                              hipStream_t stream) {
  (void)n_in; (void)out_size; (void)ws_size;

  const float* x     = (const float*)d_in[0];
  const int*   ei    = (const int*)d_in[1];   // [2, E]: row0 = src, row1 = dst
  const float* gamma = (const float*)d_in[2];

  const int N     = in_sizes[0] / CH;
  const int E     = in_sizes[1] / 2;
  const int steps = in_sizes[2] - 1;

  const int* src = ei;
  const int* dst = ei + E;

  // carve 256B-aligned workspace regions
  char* w = (char*)d_ws;
  auto carve = [&](size_t bytes) -> void* {
    void* p = (void*)w;
    w += (bytes + 255) & ~(size_t)255;
    return p;
  };
  int*   degi   = (int*)carve((size_t)N * 4);
  int*   rowptr = (int*)carve((size_t)(N + 1) * 4);
  int*   cursor = (int*)carve((size_t)N * 4);
  float* dinv   = (float*)carve((size_t)N * 4);
  int*   col    = (int*)carve((size_t)E * 4);
  float* val    = (float*)carve((size_t)E * 4);
  float* hA     = (float*)carve((size_t)N * CH * 4);
  float* hB     = (float*)carve((size_t)N * CH * 4);
  float* out    = (float*)d_out;

  const int n4  = N * CH / 4;
  const int gN  = (N + TPB - 1) / TPB;
  const int gE  = (E + TPB - 1) / TPB;
  const int g4  = (n4 + TPB - 1) / TPB;
  const int gP  = (N + NPB - 1) / NPB;
  const int gZ  = ((N + 3) / 4 + TPB - 1) / TPB;

  // ---- build transposed CSR (once per call) ----
  k_zero_i4<<<gZ, TPB, 0, stream>>>((int4*)degi,   (N + 3) / 4);
  k_zero_i4<<<gZ, TPB, 0, stream>>>((int4*)cursor, (N + 3) / 4);
  k_deg    <<<gE, TPB, 0, stream>>>(dst, degi, E);
  k_scan   <<<1, 1024, 0, stream>>>(degi, rowptr, N);
  k_dinv   <<<gN, TPB, 0, stream>>>(degi, dinv, N);
  k_fill   <<<gE, TPB, 0, stream>>>(src, dst, rowptr, cursor, dinv, col, val, E);

  // ---- out = gamma[0] * x ----
  k_scale<<<g4, TPB, 0, stream>>>((const float4*)x, (float4*)out, gamma, n4);

  // ---- 10 fused propagation steps, ping-pong h buffers ----
  const float* hc = x;
  float* bufs[2] = {hA, hB};
  for (int k = 0; k < steps; ++k) {
    float* hnext = bufs[k & 1];
    int writeH = (k + 1 < steps) ? 1 : 0;   // last step: out only
    k_prop<<<gP, TPB, 0, stream>>>(hc, hnext, out, rowptr, col, val,
                                   gamma, k + 1, N, writeH);
    hc = hnext;
  }
}